// IntrinsicRewardModule_53781580480581
// MI455X (gfx1250) — compile-verified
//
#include <hip/hip_runtime.h>

// CDNA5 / gfx1250: wave32, WMMA f32 16x16x4 (full f32 precision reduction-by-matmul).
typedef __attribute__((ext_vector_type(2))) float v2f;
typedef __attribute__((ext_vector_type(8))) float v8f;

#define LAMBDA_INT 0.01f
#define EPS_F 1e-8f

// ---------------------------------------------------------------------------
// Kernel A: r_raw[b] = mean_d (zp[b,d]-zt[b,d])^2
// One block = 16 rows (one WMMA M-tile). 4 waves split the column dimension.
// Row sums computed with V_WMMA_F32_16X16X4_F32 against a ones B-matrix:
//   - result is K-assignment invariant, so lane L (rows 0..15) and lane L+16
//     take disjoint 4-column chunks of the same row via contiguous float4 loads.
//   - C accumulator chains across the k loop in f32 (same rounding as scalar FMA).
// Loop trip count is scalar-uniform (D/32 from a kernel arg) so the backend
// emits a pure SALU loop: EXEC stays all-1s across every WMMA (ISA requirement).
// D-matrix layout: VGPR v -> M=v on lanes 0..15, M=v+8 on lanes 16..31 (any N).
// ---------------------------------------------------------------------------
__global__ __launch_bounds__(128) void rint_rows_wmma(
    const float* __restrict__ zp, const float* __restrict__ zt,
    float* __restrict__ r_raw, int D)
{
    __shared__ float tile_sum[16];

    const int tid  = threadIdx.x;
    const int lane = tid & 31;
    const int wave = tid >> 5;      // 0..3
    const int m    = lane & 15;     // row within 16-row tile
    const int half = lane >> 4;     // 0 or 1 -> which 4-col chunk of the 8-col step

    const long row = (long)blockIdx.x * 16 + m;
    const int  col0 = wave * 8 + half * 4;            // this lane's starting column
    const float* __restrict__ pbase = zp + row * (long)D + col0;
    const float* __restrict__ tbase = zt + row * (long)D + col0;

    if (tid < 16) tile_sum[tid] = 0.0f;
    __syncthreads();

    v8f c = {0.f, 0.f, 0.f, 0.f, 0.f, 0.f, 0.f, 0.f};
    v2f bones = {1.0f, 1.0f};   // B = ones (4x16): every lane holds 1.0 in both K slots

    // Each wave covers 8 columns per iteration (4 per lane-half), stride 32 across
    // 4 waves. Requires D % 32 == 0 (D = 4096 here). `iters` is wave-uniform.
    const int iters = D >> 5;
    #pragma unroll 4
    for (int i = 0; i < iters; ++i) {
        const float* pp = pbase + ((long)i << 5);
        const float* tp = tbase + ((long)i << 5);

        // Prefetch ~256 columns ahead (speculative; silently dropped at buffer end).
        __builtin_prefetch(pp + 256, 0, 3);
        __builtin_prefetch(tp + 256, 0, 3);

        const float4 p = *(const float4*)pp;
        const float4 t = *(const float4*)tp;

        float4 d;
        d.x = p.x - t.x;  d.y = p.y - t.y;
        d.z = p.z - t.z;  d.w = p.w - t.w;

        v2f a0 = {d.x, d.y};
        v2f a1 = {d.z, d.w};

        // (neg_a, A, neg_b, B, c_mod, C, reuse_a, reuse_b)
        c = __builtin_amdgcn_wmma_f32_16x16x4_f32(false, a0, false, bones,
                                                  (short)0, c, false, false);
        c = __builtin_amdgcn_wmma_f32_16x16x4_f32(false, a1, false, bones,
                                                  (short)0, c, false, false);
    }

    // N=0 column of D holds the row sums (all N columns identical since B==1).
    if (lane == 0) {
        #pragma unroll
        for (int v = 0; v < 8; ++v) atomicAdd(&tile_sum[v], c[v]);        // rows 0..7
    } else if (lane == 16) {
        #pragma unroll
        for (int v = 0; v < 8; ++v) atomicAdd(&tile_sum[v + 8], c[v]);    // rows 8..15
    }
    __syncthreads();

    if (tid < 16)
        r_raw[(long)blockIdx.x * 16 + tid] = tile_sum[tid] * (1.0f / (float)D);
}

// ---------------------------------------------------------------------------
// Block-wide f32 sum (wave32 shuffles + LDS), result broadcast to all threads.
// ---------------------------------------------------------------------------
__device__ __forceinline__ float block_sum(float x, float* red)
{
    const int lane = threadIdx.x & 31;
    const int wid  = threadIdx.x >> 5;

    #pragma unroll
    for (int off = 16; off > 0; off >>= 1) x += __shfl_down(x, off, 32);
    if (lane == 0) red[wid] = x;
    __syncthreads();

    float s;
    if (wid == 0) {
        s = (lane < (int)(blockDim.x >> 5)) ? red[lane] : 0.0f;
        #pragma unroll
        for (int off = 16; off > 0; off >>= 1) s += __shfl_down(s, off, 32);
        if (lane == 0) red[0] = s;
    }
    __syncthreads();
    s = red[0];
    __syncthreads();   // allow red[] reuse by a subsequent call
    return s;
}

// ---------------------------------------------------------------------------
// Kernel B: batch mean / centered M2 over r_raw, Chan parallel-Welford merge
// with running (count, mean, M2); emits stats = { new_mean, 1/(std+eps) }.
// Single block; r_raw (128 KB) re-read from L2 for the centered second pass.
// ---------------------------------------------------------------------------
__global__ __launch_bounds__(1024) void rint_stats(
    const float* __restrict__ r_raw, int B,
    const float* __restrict__ count_p, const float* __restrict__ mean_p,
    const float* __restrict__ M2_p, float* __restrict__ stats)
{
    __shared__ float red[32];
    const int tid = threadIdx.x;

    // Pass 1: batch mean.
    float s = 0.0f;
    for (int i = tid; i < B; i += 1024) s += r_raw[i];
    const float n      = (float)B;
    const float b_mean = block_sum(s, red) / n;

    // Pass 2: centered M2 (matches reference jnp.sum(square(r - b_mean))).
    float q = 0.0f;
    for (int i = tid; i < B; i += 1024) {
        const float d = r_raw[i] - b_mean;
        q += d * d;
    }
    const float b_M2 = block_sum(q, red);

    if (tid == 0) {
        const float count = *count_p;
        const float mean  = *mean_p;
        const float M2    = *M2_p;

        const float new_count = count + n;
        const float delta     = b_mean - mean;
        const float new_mean  = mean + delta * n / new_count;
        const float new_M2    = M2 + b_M2 + delta * delta * count * n / new_count;

        const float var = new_M2 / fmaxf(new_count - 1.0f, 1.0f);
        const float std = (new_count < 2.0f) ? 1.0f : (sqrtf(var) + EPS_F);

        stats[0] = new_mean;
        stats[1] = 1.0f / (std + EPS_F);
    }
}

// ---------------------------------------------------------------------------
// Kernel C: out = LAMBDA_INT * (r_raw - new_mean) * 1/(std+eps)
// ---------------------------------------------------------------------------
__global__ void rint_norm(const float* __restrict__ r_raw,
                          const float* __restrict__ stats,
                          float* __restrict__ out, int B)
{
    const int i = blockIdx.x * blockDim.x + threadIdx.x;
    if (i < B) out[i] = LAMBDA_INT * (r_raw[i] - stats[0]) * stats[1];
}

// ---------------------------------------------------------------------------
extern "C" void kernel_launch(void* const* d_in, const int* in_sizes, int n_in,
                              void* d_out, int out_size, void* d_ws, size_t ws_size,
                              hipStream_t stream)
{
    const float* zp    = (const float*)d_in[0];
    const float* zt    = (const float*)d_in[1];
    const float* count = (const float*)d_in[2];
    const float* mean  = (const float*)d_in[3];
    const float* M2    = (const float*)d_in[4];
    float* out = (float*)d_out;

    const int B = out_size;                       // 32768
    const int D = (int)((long)in_sizes[0] / B);   // 4096

    float* r_raw = (float*)d_ws;                  // B floats
    float* stats = r_raw + B;                     // 2 floats

    rint_rows_wmma<<<B / 16, 128, 0, stream>>>(zp, zt, r_raw, D);
    rint_stats<<<1, 1024, 0, stream>>>(r_raw, B, count, mean, M2, stats);
    rint_norm<<<(B + 255) / 256, 256, 0, stream>>>(r_raw, stats, out, B);
}